// GraphNet_37769942401472
// MI455X (gfx1250) — compile-verified
//
#include <hip/hip_runtime.h>

typedef __attribute__((ext_vector_type(2))) float v2f;
typedef __attribute__((ext_vector_type(8))) float v8f;

#define N_NODES 100000
#define IN_FEAT 32
#define EMBED   64
#define OUT_DIM 128
#define N_EDGES 1600000

// ---------------------------------------------------------------------------
// f32 tensor-core MAC: D = A(16x4 f32) * B(4x16 f32) + C, full precision.
// ---------------------------------------------------------------------------
__device__ __forceinline__ v8f wmma4(v2f a, v2f b, v8f c) {
  return __builtin_amdgcn_wmma_f32_16x16x4_f32(
      /*neg_a=*/false, a, /*neg_b=*/false, b,
      /*c_mod=*/(short)0, c, /*reuse_a=*/false, /*reuse_b=*/false);
}

// Accumulate a 16-row strip:  acc[t] += Arow[16xK] (row-scaled) @ B[K x (NT*16)]
// Arow points at &A[row0*K] (row-major, ld=K).  B row-major with ld=ldb.
// A layout: lane m = lane&15 supplies row m; K-pair base = k0 + (lane>>4)*2.
// B layout: symmetric (lane n = lane&15 supplies column n of the tile).
template <int K, int NT>
__device__ __forceinline__ void mm_accum(const float* __restrict__ Arow,
                                         float ascale,
                                         const float* __restrict__ B, int ldb,
                                         v8f* acc, int lane) {
  const int m  = lane & 15;
  const int kh = (lane >> 4) * 2;
  for (int k0 = 0; k0 < K; k0 += 4) {
    const int kb = k0 + kh;
    float2 av = *(const float2*)(Arow + m * K + kb);   // kb even -> 8B aligned
    v2f a;
    a.x = av.x * ascale;
    a.y = av.y * ascale;
#pragma unroll
    for (int t = 0; t < NT; ++t) {
      v2f b;
      const int n = t * 16 + m;
      b.x = B[kb * ldb + n];
      b.y = B[(kb + 1) * ldb + n];
      acc[t] = wmma4(a, b, acc[t]);
    }
  }
}

// Store C tiles (+bias, optional ReLU).  D row-major, ld = NT*16.
template <int NT, bool RELU>
__device__ __forceinline__ void store_tiles(float* __restrict__ D, int row0,
                                            const float* __restrict__ bias,
                                            const v8f* acc, int lane) {
  const int n  = lane & 15;
  const int mo = (lane >> 4) * 8;
#pragma unroll
  for (int t = 0; t < NT; ++t) {
    const float bv = bias[t * 16 + n];
#pragma unroll
    for (int r = 0; r < 8; ++r) {
      float v = acc[t][r] + bv;
      if (RELU) v = fmaxf(v, 0.0f);
      D[(size_t)(row0 + mo + r) * (NT * 16) + t * 16 + n] = v;
    }
  }
}

// ---------------------------------------------------------------------------
// Kernels
// ---------------------------------------------------------------------------
__global__ void zero_kernel(float* __restrict__ p, long long n) {
  long long i = (long long)blockIdx.x * blockDim.x + threadIdx.x;
  if (i < n) p[i] = 0.0f;
}

__global__ void count_kernel(const int* __restrict__ edges,
                             float* __restrict__ cnt) {
  int e = blockIdx.x * blockDim.x + threadIdx.x;
  if (e >= N_EDGES) return;
  int d = edges[N_EDGES + e];
  atomicAdd(cnt + d, 1.0f);
}

// 16 threads per edge; each moves 4 consecutive floats (coalesced gather,
// consecutive-address f32 atomics that resolve in the 192MB L2).
__global__ void scatter_kernel(const float* __restrict__ h,
                               const int* __restrict__ edges,
                               float* __restrict__ agg) {
  long long t = (long long)blockIdx.x * blockDim.x + threadIdx.x;
  if (t >= (long long)N_EDGES * 16) return;
  const int e = (int)(t >> 4);
  const int c = (int)(t & 15) * 4;
  const int s = edges[e];
  const int d = edges[N_EDGES + e];
  float4 v = *(const float4*)(h + (size_t)s * EMBED + c);
  float* ap = agg + (size_t)d * EMBED + c;
  atomicAdd(ap + 0, v.x);
  atomicAdd(ap + 1, v.y);
  atomicAdd(ap + 2, v.z);
  atomicAdd(ap + 3, v.w);
}

// h = x @ W_emb + b_emb      (16x64 strip per wave, K=32)
__global__ __launch_bounds__(128) void embed_kernel(
    const float* __restrict__ x, const float* __restrict__ W,
    const float* __restrict__ b, float* __restrict__ h) {
  const int lane  = threadIdx.x & 31;
  const int strip = blockIdx.x * 4 + (threadIdx.x >> 5);
  if (strip * 16 >= N_NODES) return;
  const int row0 = strip * 16;
  v8f acc[4] = {};
  mm_accum<IN_FEAT, 4>(x + (size_t)row0 * IN_FEAT, 1.0f, W, EMBED, acc, lane);
  store_tiles<4, false>(h, row0, b, acc, lane);
}

// hout = relu( (agg/clamp(cnt,1)) @ Wrel + h @ Wroot + bias )
// Both matmuls fused into the same WMMA accumulators; 1/cnt folded into the
// A-operand load of the first matmul.
__global__ __launch_bounds__(128) void layer_kernel(
    const float* __restrict__ h, const float* __restrict__ agg,
    const float* __restrict__ cnt, const float* __restrict__ Wrel,
    const float* __restrict__ Wroot, const float* __restrict__ bias,
    float* __restrict__ hout) {
  const int lane  = threadIdx.x & 31;
  const int strip = blockIdx.x * 4 + (threadIdx.x >> 5);
  if (strip * 16 >= N_NODES) return;
  const int row0 = strip * 16;
  const float inv = 1.0f / fmaxf(cnt[row0 + (lane & 15)], 1.0f);
  v8f acc[4] = {};
  mm_accum<EMBED, 4>(agg + (size_t)row0 * EMBED, inv, Wrel, EMBED, acc, lane);
  mm_accum<EMBED, 4>(h + (size_t)row0 * EMBED, 1.0f, Wroot, EMBED, acc, lane);
  store_tiles<4, true>(hout, row0, bias, acc, lane);
}

// out = h @ W_out + b_out    (16x128 strip per wave, 8 accumulator tiles)
__global__ __launch_bounds__(128) void out_kernel(
    const float* __restrict__ h, const float* __restrict__ W,
    const float* __restrict__ b, float* __restrict__ out) {
  const int lane  = threadIdx.x & 31;
  const int strip = blockIdx.x * 4 + (threadIdx.x >> 5);
  if (strip * 16 >= N_NODES) return;
  const int row0 = strip * 16;
  v8f acc[8] = {};
  mm_accum<EMBED, 8>(h + (size_t)row0 * EMBED, 1.0f, W, OUT_DIM, acc, lane);
  store_tiles<8, false>(out, row0, b, acc, lane);
}

// ---------------------------------------------------------------------------
extern "C" void kernel_launch(void* const* d_in, const int* in_sizes, int n_in,
                              void* d_out, int out_size, void* d_ws,
                              size_t ws_size, hipStream_t stream) {
  const float* x      = (const float*)d_in[0];
  const int*   edges  = (const int*)d_in[1];     // [2, E] (src row, dst row)
  const float* W_emb  = (const float*)d_in[2];
  const float* b_emb  = (const float*)d_in[3];
  const float* W1_rel = (const float*)d_in[4];
  const float* W1_rt  = (const float*)d_in[5];
  const float* b1     = (const float*)d_in[6];
  const float* W2_rel = (const float*)d_in[7];
  const float* W2_rt  = (const float*)d_in[8];
  const float* b2     = (const float*)d_in[9];
  const float* W_out  = (const float*)d_in[10];
  const float* b_out  = (const float*)d_in[11];
  float* out = (float*)d_out;

  // Workspace layout (floats): hA | hB | agg | cnt  (~77 MB)
  float* hA  = (float*)d_ws;
  float* hB  = hA + (size_t)N_NODES * EMBED;
  float* agg = hB + (size_t)N_NODES * EMBED;
  float* cnt = agg + (size_t)N_NODES * EMBED;

  const int strips = N_NODES / 16;               // 6250 (N divisible by 16)
  dim3 mmGrid((strips + 3) / 4), mmBlock(128);
  const long long scatT = (long long)N_EDGES * 16;
  const int scatGrid = (int)((scatT + 255) / 256);
  const long long aggN = (long long)N_NODES * EMBED;

  // Zero agg + cnt (contiguous), build degree counts once (shared by layers).
  zero_kernel<<<(int)((aggN + N_NODES + 255) / 256), 256, 0, stream>>>(
      agg, aggN + N_NODES);
  count_kernel<<<(N_EDGES + 255) / 256, 256, 0, stream>>>(edges, cnt);

  // Embed.
  embed_kernel<<<mmGrid, mmBlock, 0, stream>>>(x, W_emb, b_emb, hA);

  // Layer 1: scatter raw features, then (agg/cnt)@Wrel + h@Wroot fused.
  scatter_kernel<<<scatGrid, 256, 0, stream>>>(hA, edges, agg);
  layer_kernel<<<mmGrid, mmBlock, 0, stream>>>(hA, agg, cnt, W1_rel, W1_rt, b1,
                                               hB);

  // Layer 2.
  zero_kernel<<<(int)((aggN + 255) / 256), 256, 0, stream>>>(agg, aggN);
  scatter_kernel<<<scatGrid, 256, 0, stream>>>(hB, edges, agg);
  layer_kernel<<<mmGrid, mmBlock, 0, stream>>>(hB, agg, cnt, W2_rel, W2_rt, b2,
                                               hA);

  // Output projection.
  out_kernel<<<mmGrid, mmBlock, 0, stream>>>(hA, W_out, b_out, out);
}